// GCN_23759759082168
// MI455X (gfx1250) — compile-verified
//
#include <hip/hip_runtime.h>
#include <hip/hip_bf16.h>

#define USER_NUM 50000
#define ITEM_NUM 50000
#define NNODES   (USER_NUM + ITEM_NUM)   // 100000
#define DIM      256
#define NEDGE    1000000

typedef __attribute__((ext_vector_type(2))) float v2f;
typedef __attribute__((ext_vector_type(8))) float v8f;

// ---------------------------------------------------------------------------
// SPMM: out[rows[e], :] += vals[e] * feat[cols[e], :]
// One wave (32 lanes) per edge; each lane handles 8 contiguous floats.
// Row gather is a fully coalesced 1KB read (hits L2: table is 102MB < 192MB L2).
// Scatter is 8 fire-and-forget global_atomic_add_f32 per lane (STOREcnt path).
// ---------------------------------------------------------------------------
__global__ __launch_bounds__(256) void spmm_kernel(
    const int* __restrict__ rows, const int* __restrict__ cols,
    const float* __restrict__ vals, const float* __restrict__ feat,
    float* __restrict__ out, int num_edges)
{
    const int gtid = blockIdx.x * blockDim.x + threadIdx.x;
    const int e    = gtid >> 5;
    const int lane = threadIdx.x & 31;
    if (e >= num_edges) return;

    const int   r = rows[e];
    const int   c = cols[e];
    const float v = vals[e];

    const float4* src = (const float4*)(feat + (size_t)c * DIM) + lane * 2;
    float4 f0 = src[0];
    float4 f1 = src[1];

    float* dst = out + (size_t)r * DIM + lane * 8;
    atomicAdd(dst + 0, v * f0.x);
    atomicAdd(dst + 1, v * f0.y);
    atomicAdd(dst + 2, v * f0.z);
    atomicAdd(dst + 3, v * f0.w);
    atomicAdd(dst + 4, v * f1.x);
    atomicAdd(dst + 5, v * f1.y);
    atomicAdd(dst + 6, v * f1.z);
    atomicAdd(dst + 7, v * f1.w);
}

// ---------------------------------------------------------------------------
// GEMM: out = relu?(X @ W^T + bias), X:[N,256], W:[256,256] row-major.
// fp32 WMMA 16x16x4. One wave per 16(M) x 64(N) tile: A frag reused across
// 4 B frags each K-step; K=256 -> 64 steps.
//
// A 16x4 f32 layout (ISA 7.12.2): lanes0-15 row m hold K={k0,k0+1} in
// VGPR0/1, lanes16-31 hold K={k0+2,k0+3}  -> float2 load at k0+2*(lane>>4).
// B 4x16 mirrors it (row-of-B striped across lanes); B[k][n]=W[n][k] so the
// per-lane float2 comes from W row n = colbase + (lane&15), contiguous in K.
// C/D: VGPR v -> M=v (lanes0-15) / M=v+8 (lanes16-31), N=lane&15.
// ---------------------------------------------------------------------------
__global__ __launch_bounds__(32) void gemm_bias_act_kernel(
    const float* __restrict__ X, const float* __restrict__ W,
    const float* __restrict__ bias, float* __restrict__ out, int do_relu)
{
    const int lane    = threadIdx.x;        // 0..31, wave32
    const int rowbase = blockIdx.x * 16;    // M tile
    const int stripe  = blockIdx.y * 64;    // N stripe (4 x 16 tiles)
    const int m       = lane & 15;
    const int koff    = (lane >> 4) << 1;   // 0 or 2

    v8f acc[4];
#pragma unroll
    for (int t = 0; t < 4; ++t) acc[t] = (v8f)(0.0f);

    const float* xrow = X + (size_t)(rowbase + m) * DIM;
    const float* wrow0 = W + (size_t)(stripe + m) * DIM;

    for (int k0 = 0; k0 < DIM; k0 += 4) {
        v2f a = *(const v2f*)(xrow + k0 + koff);
#pragma unroll
        for (int t = 0; t < 4; ++t) {
            v2f b = *(const v2f*)(wrow0 + (size_t)t * 16 * DIM + k0 + koff);
            // 8 args: (neg_a, A, neg_b, B, c_mod, C, reuse_a, reuse_b)
            acc[t] = __builtin_amdgcn_wmma_f32_16x16x4_f32(
                false, a, false, b, (short)0, acc[t], false, false);
        }
    }

    const int rhi = (lane >> 4) * 8;
#pragma unroll
    for (int t = 0; t < 4; ++t) {
        const int col = stripe + t * 16 + m;
        const float bv = bias[col];
#pragma unroll
        for (int v = 0; v < 8; ++v) {
            const int row = rowbase + v + rhi;
            float val = acc[t][v] + bv;
            if (do_relu) val = fmaxf(val, 0.0f);
            out[(size_t)row * DIM + col] = val;
        }
    }
}

extern "C" void kernel_launch(void* const* d_in, const int* in_sizes, int n_in,
                              void* d_out, int out_size, void* d_ws, size_t ws_size,
                              hipStream_t stream) {
    const int*   adj_rows = (const int*)d_in[0];
    const int*   adj_cols = (const int*)d_in[1];
    const float* adj_vals = (const float*)d_in[2];
    const float* user_emb = (const float*)d_in[3];
    const float* item_emb = (const float*)d_in[4];
    const float* W        = (const float*)d_in[5];   // [3,256,256]
    const float* B        = (const float*)d_in[6];   // [3,256]

    float* feat = (float*)d_out;  // buffer A: [N,256] (also final output)
    float* acc  = (float*)d_ws;   // buffer B: [N,256] SPMM accumulator

    const size_t featBytes = (size_t)NNODES * DIM * sizeof(float);

    // concat(user_emb, item_emb) -> feat
    hipMemcpyAsync(feat, user_emb, (size_t)USER_NUM * DIM * sizeof(float),
                   hipMemcpyDeviceToDevice, stream);
    hipMemcpyAsync(feat + (size_t)USER_NUM * DIM, item_emb,
                   (size_t)ITEM_NUM * DIM * sizeof(float),
                   hipMemcpyDeviceToDevice, stream);

    const int spmm_blocks = (NEDGE * 32 + 255) / 256;       // 1 wave per edge
    dim3 gemm_grid(NNODES / 16, 4);                          // 6250 x 4 waves

    for (int l = 0; l < 3; ++l) {
        hipMemsetAsync(acc, 0, featBytes, stream);
        spmm_kernel<<<spmm_blocks, 256, 0, stream>>>(
            adj_rows, adj_cols, adj_vals, feat, acc, NEDGE);
        gemm_bias_act_kernel<<<gemm_grid, 32, 0, stream>>>(
            acc, W + (size_t)l * DIM * DIM, B + (size_t)l * DIM,
            feat, (l < 2) ? 1 : 0);
    }
}